// AudioAttNet_317827580578
// MI455X (gfx1250) — compile-verified
//
#include <hip/hip_runtime.h>
#include <hip/hip_bf16.h>
#include <math.h>

// ---------------------------------------------------------------------------
// MI455X (gfx1250) implementation.
// Roofline: dominated by Wl (256MB) + x (2x64MB) reads -> ~17us floor at
// 23.3 TB/s HBM; FLOPs (~1.8G) are negligible -> stay f32 end-to-end.
// conv1 (Cout=16, Cin=2048, k=3, L=8192) runs on V_WMMA_F32_16X16X4_F32
// (wave32, one 16-position tile per wave) with:
//   - A (weights) pre-packed into WMMA lane layout -> coalesced b64 loads
//   - B (x rows)  loaded as b64 with clamped pointers + {0,1} lane mask
//     (no EXEC-masked branches in the hot loop)
// ---------------------------------------------------------------------------

#define SEQ 8192
#define DIM 2048

typedef __attribute__((ext_vector_type(2))) float v2f;
typedef __attribute__((ext_vector_type(8))) float v8f;

__device__ __forceinline__ float leaky(float v) {
    return v >= 0.0f ? v : 0.02f * v;
}

// ---------------------------------------------------------------------------
// Pack w1 [16][2048][3] into WMMA A-operand lane layout:
//   wp[((k*512 + g)*32 + lane)*2 + j] = w1[m][c][k]
//     m = lane&15,  c = 4*g + 2*(lane>>4) + j
// so each lane's A fragment for K-step g is one contiguous 8-byte load.
// ---------------------------------------------------------------------------
__global__ __launch_bounds__(256)
void pack_w1_kernel(const float* __restrict__ w1, float* __restrict__ wp)
{
    const int idx = blockIdx.x * 256 + threadIdx.x;     // 3*512*64 = 98304
    if (idx >= 3 * 512 * 64) return;
    const int j    = idx & 1;
    const int lane = (idx >> 1) & 31;
    const int g    = (idx >> 6) & 511;
    const int k    = idx >> 15;                          // 0..2
    const int m    = lane & 15;
    const int c    = 4 * g + 2 * (lane >> 4) + j;
    wp[idx] = w1[(m * DIM + c) * 3 + k];
}

// ---------------------------------------------------------------------------
// conv1: y1[m][l] = leaky( b1[m] + sum_k sum_ci w1[m][ci][k]*x[l+k-1][ci] )
// WMMA f32 16x16x4 mapping:
//   A (16x4):  lane L: M = L&15; VGPR0/1 hold K = 2*(L>>4) + {0,1}
//   B (4x16):  lane L: N = L&15; VGPR0/1 hold K = 2*(L>>4) + {0,1}
//   C/D:       VGPR r, lane L -> M = r + 8*(L>>4), N = L&15
// ---------------------------------------------------------------------------
__global__ __launch_bounds__(32)
void conv1_wmma_kernel(const float* __restrict__ x,   // [SEQ][DIM]
                       const float* __restrict__ wp,  // packed A, [3][512][32][2]
                       const float* __restrict__ b1,  // [16]
                       float* __restrict__ y1)        // [16][SEQ]
{
    const int tile = blockIdx.x;       // 512 tiles of 16 positions
    const int lane = threadIdx.x;      // 0..31
    const int half = lane >> 4;        // 0 | 1
    const int lidx = lane & 15;
    const int base = tile * 16;

    v8f acc = {};

    #pragma unroll
    for (int k = 0; k < 3; ++k) {
        // B-lane: position (column n = lidx) shifted by tap; clamp + mask
        const int  row    = base + lidx + (k - 1);
        const bool inb    = (row >= 0) && (row < SEQ);
        const float bscale = inb ? 1.0f : 0.0f;
        const float* xr = x + (size_t)(inb ? row : 0) * DIM + half * 2;
        const float* wr = wp + (size_t)k * (512 * 64) + lane * 2;

        #pragma unroll 4
        for (int g = 0; g < 512; ++g) {
            v2f a = *reinterpret_cast<const v2f*>(wr + g * 64);  // coalesced b64
            v2f b = *reinterpret_cast<const v2f*>(xr + g * 4);   // b64
            b.x *= bscale;
            b.y *= bscale;
            acc = __builtin_amdgcn_wmma_f32_16x16x4_f32(
                /*neg_a=*/false, a, /*neg_b=*/false, b,
                /*c_mod=*/(short)0, acc, /*reuse_a=*/false, /*reuse_b=*/false);
        }
    }

    #pragma unroll
    for (int r = 0; r < 8; ++r) {
        const int m   = r + half * 8;
        const int col = base + lidx;
        y1[(size_t)m * SEQ + col] = leaky(acc[r] + b1[m]);
    }
}

// ---------------------------------------------------------------------------
// Small conv stages (Cin<=16): one thread per output position.
// ---------------------------------------------------------------------------
template <int CIN, int COUT>
__global__ __launch_bounds__(256)
void conv_small_kernel(const float* __restrict__ in,  // [CIN][SEQ]
                       const float* __restrict__ w,   // [COUT][CIN][3]
                       const float* __restrict__ b,   // [COUT]
                       float* __restrict__ out)       // [COUT][SEQ]
{
    const int l = blockIdx.x * 256 + threadIdx.x;
    if (l >= SEQ) return;
    #pragma unroll
    for (int co = 0; co < COUT; ++co) {
        float s = b[co];
        #pragma unroll
        for (int k = 0; k < 3; ++k) {
            const int r = l + k - 1;
            if (r < 0 || r >= SEQ) continue;
            #pragma unroll
            for (int ci = 0; ci < CIN; ++ci)
                s = fmaf(w[(co * CIN + ci) * 3 + k], in[ci * SEQ + r], s);
        }
        out[co * SEQ + l] = leaky(s);
    }
}

// ---------------------------------------------------------------------------
// logits[row] = dot(Wl[row,:], v) + bl[row].  One 256-thread block per row.
// This is the 256MB stream that sets the kernel's runtime floor.
// ---------------------------------------------------------------------------
__global__ __launch_bounds__(256)
void matvec_kernel(const float* __restrict__ Wl,   // [SEQ][SEQ]
                   const float* __restrict__ v,    // [SEQ]
                   const float* __restrict__ bl,   // [SEQ]
                   float* __restrict__ logits)     // [SEQ]
{
    const int row = blockIdx.x;
    const float* w = Wl + (size_t)row * SEQ;
    float s = 0.0f;
    for (int j = threadIdx.x * 4; j < SEQ; j += 256 * 4) {
        __builtin_prefetch(w + j + 2048, 0, 1);   // global_prefetch_b8
        const float4 wv = *reinterpret_cast<const float4*>(w + j);
        const float4 vv = *reinterpret_cast<const float4*>(v + j);
        s = fmaf(wv.x, vv.x, s);
        s = fmaf(wv.y, vv.y, s);
        s = fmaf(wv.z, vv.z, s);
        s = fmaf(wv.w, vv.w, s);
    }
    #pragma unroll
    for (int off = 16; off > 0; off >>= 1) s += __shfl_down(s, off, 32);
    __shared__ float partial[8];
    if ((threadIdx.x & 31) == 0) partial[threadIdx.x >> 5] = s;
    __syncthreads();
    if (threadIdx.x == 0) {
        float t = 0.0f;
        #pragma unroll
        for (int i = 0; i < 8; ++i) t += partial[i];
        logits[row] = t + bl[row];
    }
}

// ---------------------------------------------------------------------------
// Single-block softmax over 8192 logits -> att.
// ---------------------------------------------------------------------------
__global__ __launch_bounds__(1024)
void softmax_kernel(const float* __restrict__ logits, float* __restrict__ att)
{
    __shared__ float red[32];
    const int tid = threadIdx.x;

    // ---- max ----
    float m = -INFINITY;
    for (int i = tid; i < SEQ; i += 1024) m = fmaxf(m, logits[i]);
    #pragma unroll
    for (int off = 16; off > 0; off >>= 1) m = fmaxf(m, __shfl_down(m, off, 32));
    if ((tid & 31) == 0) red[tid >> 5] = m;
    __syncthreads();
    if (tid < 32) {
        float t = red[tid];
        #pragma unroll
        for (int off = 16; off > 0; off >>= 1) t = fmaxf(t, __shfl_down(t, off, 32));
        if (tid == 0) red[0] = t;
    }
    __syncthreads();
    m = red[0];
    __syncthreads();

    // ---- exp + sum ----
    float s = 0.0f;
    for (int i = tid; i < SEQ; i += 1024) {
        const float e = expf(logits[i] - m);
        att[i] = e;
        s += e;
    }
    #pragma unroll
    for (int off = 16; off > 0; off >>= 1) s += __shfl_down(s, off, 32);
    if ((tid & 31) == 0) red[tid >> 5] = s;
    __syncthreads();
    if (tid < 32) {
        float t = red[tid];
        #pragma unroll
        for (int off = 16; off > 0; off >>= 1) t += __shfl_down(t, off, 32);
        if (tid == 0) red[0] = t;
    }
    __syncthreads();

    const float inv = 1.0f / red[0];
    for (int i = tid; i < SEQ; i += 1024) att[i] *= inv;
}

// ---------------------------------------------------------------------------
// out[d] = sum_i att[i] * x[i][d], split over 32 row-chunks for MLP
// (2048 waves in flight instead of 64), then reduced deterministically.
// ---------------------------------------------------------------------------
__global__ __launch_bounds__(256)
void wsum_partial_kernel(const float* __restrict__ x,    // [SEQ][DIM]
                         const float* __restrict__ att,  // [SEQ]
                         float* __restrict__ partial)    // [32][DIM]
{
    const int d  = blockIdx.x * 256 + threadIdx.x;  // grid.x = 8
    const int c  = blockIdx.y;                      // grid.y = 32
    const int i0 = c * (SEQ / 32);
    float s = 0.0f;
    #pragma unroll 4
    for (int i = i0; i < i0 + SEQ / 32; ++i)
        s = fmaf(att[i], x[(size_t)i * DIM + d], s);
    partial[c * DIM + d] = s;
}

__global__ __launch_bounds__(256)
void wsum_final_kernel(const float* __restrict__ partial,  // [32][DIM]
                       float* __restrict__ out)            // [DIM]
{
    const int d = blockIdx.x * 256 + threadIdx.x;
    float s = 0.0f;
    #pragma unroll
    for (int c = 0; c < 32; ++c) s += partial[c * DIM + d];
    out[d] = s;
}

// ---------------------------------------------------------------------------
// Launch
// Inputs: 0:x 1:w1 2:b1 3:w2 4:b2 5:w3 6:b3 7:w4 8:b4 9:w5 10:b5 11:Wl 12:bl
// ---------------------------------------------------------------------------
extern "C" void kernel_launch(void* const* d_in, const int* in_sizes, int n_in,
                              void* d_out, int out_size, void* d_ws, size_t ws_size,
                              hipStream_t stream) {
    (void)in_sizes; (void)n_in; (void)out_size; (void)ws_size;

    const float* x  = (const float*)d_in[0];
    const float* w1 = (const float*)d_in[1];
    const float* b1 = (const float*)d_in[2];
    const float* w2 = (const float*)d_in[3];
    const float* b2 = (const float*)d_in[4];
    const float* w3 = (const float*)d_in[5];
    const float* b3 = (const float*)d_in[6];
    const float* w4 = (const float*)d_in[7];
    const float* b4 = (const float*)d_in[8];
    const float* w5 = (const float*)d_in[9];
    const float* b5 = (const float*)d_in[10];
    const float* Wl = (const float*)d_in[11];
    const float* bl = (const float*)d_in[12];
    float* out = (float*)d_out;

    // workspace layout (floats)
    float* ws      = (float*)d_ws;
    float* y1      = ws;                     // 16*SEQ  = 131072
    float* y2      = y1 + 16 * SEQ;          //  8*SEQ  =  65536
    float* y3      = y2 + 8 * SEQ;           //  4*SEQ  =  32768
    float* y4      = y3 + 4 * SEQ;           //  2*SEQ  =  16384
    float* y5      = y4 + 2 * SEQ;           //  1*SEQ  =   8192
    float* logits  = y5 + SEQ;               //            8192
    float* att     = logits + SEQ;           //            8192
    float* wpack   = att + SEQ;              // 3*512*64 = 98304
    float* partial = wpack + 3 * 512 * 64;   // 32*DIM   = 65536
    // total ~1.74 MB

    pack_w1_kernel<<<(3 * 512 * 64) / 256, 256, 0, stream>>>(w1, wpack);
    conv1_wmma_kernel<<<SEQ / 16, 32, 0, stream>>>(x, wpack, b1, y1);
    conv_small_kernel<16, 8><<<SEQ / 256, 256, 0, stream>>>(y1, w2, b2, y2);
    conv_small_kernel<8, 4><<<SEQ / 256, 256, 0, stream>>>(y2, w3, b3, y3);
    conv_small_kernel<4, 2><<<SEQ / 256, 256, 0, stream>>>(y3, w4, b4, y4);
    conv_small_kernel<2, 1><<<SEQ / 256, 256, 0, stream>>>(y4, w5, b5, y5);
    matvec_kernel<<<SEQ, 256, 0, stream>>>(Wl, y5, bl, logits);
    softmax_kernel<<<1, 1024, 0, stream>>>(logits, att);
    wsum_partial_kernel<<<dim3(DIM / 256, 32), 256, 0, stream>>>(x, att, partial);
    wsum_final_kernel<<<DIM / 256, 256, 0, stream>>>(partial, out);
}